// GNN_14955076124922
// MI455X (gfx1250) — compile-verified
//
#include <hip/hip_runtime.h>
#include <hip/hip_bf16.h>

typedef __attribute__((ext_vector_type(2))) float v2f;
typedef __attribute__((ext_vector_type(8))) float v8f;

#define HID 128

// ---------------------------------------------------------------------------
// GEMM + bias + ReLU:  C[N x 128] = relu(A @ W + b)
//   A is [N x K] (CONCAT=false) or concat(A0[N x 128], A1[N x 128]) (CONCAT=true, K=256)
//   W is [K x 128] row-major, b is [128]
// Block: 256 threads = 8 waves; block tile = 32 rows x 128 cols.
// Each wave: two 16x16 M-tiles sharing one B fragment per K-step
//   -> 2 WMMAs per W-fragment load, two independent accumulator chains.
// A tile staged in LDS (padded stride -> conflict-free column access).
// ---------------------------------------------------------------------------
template<int K, bool CONCAT>
__global__ __launch_bounds__(256)
void gemm_relu(const float* __restrict__ A0,
               const float* __restrict__ A1,
               const float* __restrict__ W,
               const float* __restrict__ bias,
               float* __restrict__ C)
{
    constexpr int ROWS = 32;
    constexpr int LDK  = K + 4;                // pad: rows hit distinct banks
    __shared__ float As[ROWS * LDK];

    const int tid = threadIdx.x;
    const int m0  = blockIdx.x * ROWS;

    // Cooperative load of the 32 x K A-tile (float4 per thread, coalesced rows)
    constexpr int NV = ROWS * K / 4;           // float4 count (multiple of 256)
    #pragma unroll
    for (int i = tid; i < NV; i += 256) {
        const int row = (i * 4) / K;
        const int col = (i * 4) % K;
        const float* src;
        if (CONCAT && col >= 128)
            src = A1 + (size_t)(m0 + row) * 128 + (col - 128);
        else
            src = A0 + (size_t)(m0 + row) * (CONCAT ? 128 : K) + col;
        const float4 v = *(const float4*)src;
        float* d = &As[row * LDK + col];
        d[0] = v.x; d[1] = v.y; d[2] = v.z; d[3] = v.w;
    }
    __syncthreads();

    const int lane = tid & 31;
    const int wid  = tid >> 5;
    const int n0   = wid * 16;                  // this wave's 16 output cols
    const int lm   = lane & 15;                 // M (for A) / N (for B,C,D)
    const int kh   = (lane >> 4) << 1;          // K base: 0 or 2

    v8f acc0 = {};
    v8f acc1 = {};
    const float* __restrict__ arow0 = As + lm * LDK;
    const float* __restrict__ arow1 = As + (lm + 16) * LDK;
    const float* __restrict__ wcol  = W + n0 + lm;

    #pragma unroll 8
    for (int k = 0; k < K; k += 4) {
        // B fragment: W[k+kh][n0+lm], W[k+kh+1][n0+lm] (shared by both M-tiles)
        v2f b;
        b.x = wcol[(k + kh)     * HID];
        b.y = wcol[(k + kh + 1) * HID];
        // A fragments: rows lm and lm+16 of the LDS tile (8B-aligned pairs)
        v2f a0 = *(const v2f*)(arow0 + k + kh);
        v2f a1 = *(const v2f*)(arow1 + k + kh);
        acc0 = __builtin_amdgcn_wmma_f32_16x16x4_f32(
                   false, a0, false, b, (short)0, acc0, false, false);
        acc1 = __builtin_amdgcn_wmma_f32_16x16x4_f32(
                   false, a1, false, b, (short)0, acc1, false, false);
    }

    const float bv     = bias[n0 + lm];
    const int   rbase0 = m0 + ((lane >> 4) << 3);
    #pragma unroll
    for (int i = 0; i < 8; ++i) {
        float v = acc0[i] + bv;
        v = v > 0.f ? v : 0.f;
        C[(size_t)(rbase0 + i) * HID + n0 + lm] = v;
    }
    #pragma unroll
    for (int i = 0; i < 8; ++i) {
        float v = acc1[i] + bv;
        v = v > 0.f ? v : 0.f;
        C[(size_t)(rbase0 + 16 + i) * HID + n0 + lm] = v;
    }
}

// ---------------------------------------------------------------------------
// Scatter-add: one wave per edge. 512B coalesced row gather (float4/lane),
// 4 hardware float atomics per lane into the destination row.
// ---------------------------------------------------------------------------
__global__ __launch_bounds__(256)
void scatter_add(const float* __restrict__ emb,
                 const int* __restrict__ src,
                 const int* __restrict__ dst,
                 float* __restrict__ aggr, int n_edges)
{
    const int e    = (int)((blockIdx.x * blockDim.x + threadIdx.x) >> 5);
    const int lane = threadIdx.x & 31;
    if (e >= n_edges) return;
    const int s = src[e];
    const int d = dst[e];
    const float4 v = *(const float4*)(emb + (size_t)s * HID + lane * 4);
    float* p = aggr + (size_t)d * HID + lane * 4;
    unsafeAtomicAdd(p + 0, v.x);
    unsafeAtomicAdd(p + 1, v.y);
    unsafeAtomicAdd(p + 2, v.z);
    unsafeAtomicAdd(p + 3, v.w);
}

// ---------------------------------------------------------------------------
// Column sums of X[N x 128] -> cs[128] (per-block partials, then atomics)
// ---------------------------------------------------------------------------
__global__ __launch_bounds__(128)
void colsum_kernel(const float* __restrict__ X, float* __restrict__ cs, int n_rows)
{
    const int col = threadIdx.x;
    float s = 0.f;
    for (int r = blockIdx.x; r < n_rows; r += gridDim.x)
        s += X[(size_t)r * HID + col];
    unsafeAtomicAdd(&cs[col], s);
}

// out[16] = cs @ Wc + bc
__global__ void final_kernel(const float* __restrict__ cs,
                             const float* __restrict__ Wc,
                             const float* __restrict__ bc,
                             float* __restrict__ out)
{
    const int o = threadIdx.x;
    if (o < 16) {
        float s = bc[o];
        #pragma unroll 8
        for (int h = 0; h < HID; ++h) s += cs[h] * Wc[h * 16 + o];
        out[o] = s;
    }
}

// ---------------------------------------------------------------------------
extern "C" void kernel_launch(void* const* d_in, const int* in_sizes, int n_in,
                              void* d_out, int out_size, void* d_ws, size_t ws_size,
                              hipStream_t stream) {
    const float* state    = (const float*)d_in[0];
    const float* internal = (const float*)d_in[1];
    const int*   s_src    = (const int*)d_in[2];
    const int*   s_dst    = (const int*)d_in[3];
    const int*   i_src    = (const int*)d_in[4];
    const int*   i_dst    = (const int*)d_in[5];
    const float* W1  = (const float*)d_in[6];
    const float* b1  = (const float*)d_in[7];
    const float* W2  = (const float*)d_in[8];
    const float* b2  = (const float*)d_in[9];
    const float* Wu1 = (const float*)d_in[10];
    const float* bu1 = (const float*)d_in[11];
    const float* Wu2 = (const float*)d_in[12];
    const float* bu2 = (const float*)d_in[13];
    const float* Wc  = (const float*)d_in[14];
    const float* bc  = (const float*)d_in[15];

    const int    N  = in_sizes[0] / 64;       // 100000
    const int    E  = in_sizes[2];            // 1600000
    const size_t NH = (size_t)N * HID;

    float* hu1  = (float*)d_out;              // [N,128] output 0
    float* hu2r = (float*)d_out + NH;         // [N,128] output 1 (also scratch)
    float* outv = (float*)d_out + 2 * NH;     // [16]    output 2

    float* B0 = (float*)d_ws;                 // ie, later ie2
    float* B1 = B0 + NH;                      // aggr1, later se2
    float* CS = B1 + NH;                      // [128] column sums

    const dim3 blk(256);
    const int  mblocks = N / 32;              // 3125 (32-row tiles)
    const int  eblocks = (E + 7) / 8;         // 8 edges (waves) per block

    // Layer 1 embeds: se -> hu2 region (temp), ie -> B0
    gemm_relu<64, false><<<mblocks, blk, 0, stream>>>(state,    nullptr, W1, b1, hu2r);
    gemm_relu<64, false><<<mblocks, blk, 0, stream>>>(internal, nullptr, W1, b1, B0);

    // aggr1 = segsum(se[s_src]->s_dst) + segsum(ie[i_src]->i_dst)  -> B1
    hipMemsetAsync(B1, 0, NH * sizeof(float), stream);
    scatter_add<<<eblocks, blk, 0, stream>>>(hu2r, s_src, s_dst, B1, E);
    scatter_add<<<eblocks, blk, 0, stream>>>(B0,   i_src, i_dst, B1, E);

    // hu1 = relu([aggr1, ie] @ Wu1 + bu1)
    gemm_relu<256, true><<<mblocks, blk, 0, stream>>>(B1, B0, Wu1, bu1, hu1);

    // Layer 2 embeds: se2 = relu(se@W2) -> B1 ; ie2 = relu(hu1@W2) -> B0
    gemm_relu<128, false><<<mblocks, blk, 0, stream>>>(hu2r, nullptr, W2, b2, B1);
    gemm_relu<128, false><<<mblocks, blk, 0, stream>>>(hu1,  nullptr, W2, b2, B0);

    // aggr2 -> hu2 region (se now dead)
    hipMemsetAsync(hu2r, 0, NH * sizeof(float), stream);
    scatter_add<<<eblocks, blk, 0, stream>>>(B1, s_src, s_dst, hu2r, E);
    scatter_add<<<eblocks, blk, 0, stream>>>(B0, i_src, i_dst, hu2r, E);

    // hu2 = relu([aggr2, ie2] @ Wu2 + bu2)  (in place: block-local rows staged
    // through LDS before being overwritten)
    gemm_relu<256, true><<<mblocks, blk, 0, stream>>>(hu2r, B0, Wu2, bu2, hu2r);

    // out = hu2.sum(0) @ Wc + bc
    hipMemsetAsync(CS, 0, HID * sizeof(float), stream);
    colsum_kernel<<<512, 128, 0, stream>>>(hu2r, CS, N);
    final_kernel<<<1, 32, 0, stream>>>(CS, Wc, bc, outv);
}